// ScaledDotProductAttention_70368744178119
// MI455X (gfx1250) — compile-verified
//
#include <hip/hip_runtime.h>

// ----------------------------------------------------------------------------
// Scaled dot-product attention, fp32, returning (output, attn, log_attn).
// MI455X / gfx1250: wave32, V_WMMA_F32_16X16X4_F32 for both GEMMs, scores
// tile kept in the 320KB-WGP LDS. Memory-bound (~740MB traffic vs 34 GFLOP),
// so fp32 WMMA is the right precision: it exceeds the ~1.1 PFLOP/s needed to
// keep HBM (23.3 TB/s) saturated while matching the fp32 reference exactly.
// ----------------------------------------------------------------------------

#define B_    64
#define LQ_   1024
#define LK_   1024
#define D_    128
#define QT    32          // q-rows per workgroup
#define WAVES 8
#define TPB   256

typedef __attribute__((ext_vector_type(2))) float v2f;
typedef __attribute__((ext_vector_type(8))) float v8f;

__device__ __forceinline__ v8f wmma_f32_4(v2f a, v2f b, v8f c) {
  // D = A(16x4) * B(4x16) + C(16x16), fp32, wave32
  return __builtin_amdgcn_wmma_f32_16x16x4_f32(
      /*neg_a=*/false, a, /*neg_b=*/false, b,
      /*c_mod=*/(short)0, c, /*reuse_a=*/false, /*reuse_b=*/false);
}

__global__ __launch_bounds__(TPB)
void attn_fused_kernel(const float* __restrict__ qg,
                       const float* __restrict__ kg,
                       const float* __restrict__ vg,
                       const unsigned char* __restrict__ maskg,
                       float* __restrict__ outO,
                       float* __restrict__ outA,
                       float* __restrict__ outLA)
{
  extern __shared__ float smem[];
  float* sS    = smem;                  // QT*LK scores -> probs (128KB)
  float* sQ    = sS + QT * LK_;         // QT*D q tile (16KB)
  float* sRed  = sQ + QT * D_;          // 32 rows * 8 groups * 4 stats
  float* sLogZ = sRed + QT * 8 * 4;     // 32
  float* sMm   = sLogZ + QT;            // 32
  float* sLinv = sMm + QT;              // 32

  const int tid  = threadIdx.x;
  const int wave = tid >> 5;
  const int lane = tid & 31;
  const int half = lane >> 4;           // 0: lanes 0-15, 1: lanes 16-31
  const int l15  = lane & 15;

  const int b  = blockIdx.x >> 5;       // LQ/QT = 32 q-tiles per batch
  const int q0 = (blockIdx.x & 31) * QT;

  const float invT = 0.08838834764831845f; // 1/sqrt(128)

  // ---- Phase 0: stage Q tile into LDS (vectorized, coalesced) -------------
  {
    const float4* src = (const float4*)(qg + ((size_t)b * LQ_ + q0) * D_);
    float4* dst = (float4*)sQ;
    for (int i = tid; i < QT * D_ / 4; i += TPB) dst[i] = src[i];
  }
  __syncthreads();

  // ---- Phase 1: S = Q K^T / sqrt(d) via WMMA f32 16x16x4 ------------------
  // Wave w owns key-column tiles n0 = (w + 8*c)*16; one B operand feeds both
  // 16-row halves of the Q tile (2 WMMAs per B load).
  {
    const float* kb = kg + (size_t)b * LK_ * D_;
    for (int cti = 0; cti < (LK_ / 16) / WAVES; ++cti) {
      const int n0 = (wave + cti * WAVES) * 16;
      v8f acc0 = {};   // rows 0..15
      v8f acc1 = {};   // rows 16..31
      const float* krow  = kb + (size_t)(n0 + l15) * D_ + half * 2;
      const float* qrow0 = sQ + (size_t)l15 * D_ + half * 2;
      const float* qrow1 = sQ + (size_t)(16 + l15) * D_ + half * 2;
#pragma unroll
      for (int kk = 0; kk < D_; kk += 4) {
        v2f bv = *(const v2f*)(krow + kk);    // B[k..k+1][n] per half
        v2f a0 = *(const v2f*)(qrow0 + kk);   // A[m][k..k+1]
        v2f a1 = *(const v2f*)(qrow1 + kk);
        acc0 = wmma_f32_4(a0, bv, acc0);
        acc1 = wmma_f32_4(a1, bv, acc1);
      }
      // C layout: VGPR r -> M = r + 8*half, N = l15
#pragma unroll
      for (int r = 0; r < 8; ++r) {
        const int row = r + half * 8;
        sS[(size_t)row * LK_ + n0 + l15]        = acc0[r] * invT;
        sS[(size_t)(row + 16) * LK_ + n0 + l15] = acc1[r] * invT;
      }
    }
  }
  __syncthreads();

  // ---- Phase 2: per-row online (max,sum) stats, 8 threads/row -------------
  const int rrow = tid >> 3;   // 0..31
  const int g    = tid & 7;    // 0..7 : columns g*128 .. g*128+127
  const unsigned int* mrow =
      (const unsigned int*)(maskg + ((size_t)b * LQ_ + q0 + rrow) * LK_ + g * 128);
  float* srow = sS + (size_t)rrow * LK_ + g * 128;
  {
    float mU = -3.4e38f, lU = 0.f, mM = -3.4e38f, lM = 0.f;
    for (int i = 0; i < 32; ++i) {
      const unsigned int m4 = mrow[i];
#pragma unroll
      for (int j = 0; j < 4; ++j) {
        const float s = srow[i * 4 + j];
        if (s > mU) { lU *= __expf(mU - s); mU = s; }
        lU += __expf(s - mU);
        const float sm = ((m4 >> (8 * j)) & 0xFFu) ? -1e9f : s;
        if (sm > mM) { lM *= __expf(mM - sm); mM = sm; }
        lM += __expf(sm - mM);
      }
    }
    sRed[(rrow * 8 + g) * 4 + 0] = mU;
    sRed[(rrow * 8 + g) * 4 + 1] = lU;
    sRed[(rrow * 8 + g) * 4 + 2] = mM;
    sRed[(rrow * 8 + g) * 4 + 3] = lM;
  }
  __syncthreads();
  if (tid < QT) {
    float Mu = -3.4e38f, Lu = 0.f, Mm = -3.4e38f, Lm = 0.f;
    for (int gg = 0; gg < 8; ++gg) {
      const float mu = sRed[(tid * 8 + gg) * 4 + 0];
      const float lu = sRed[(tid * 8 + gg) * 4 + 1];
      if (mu > Mu) { Lu *= __expf(Mu - mu); Mu = mu; }
      Lu += lu * __expf(mu - Mu);
      const float mm = sRed[(tid * 8 + gg) * 4 + 2];
      const float lm = sRed[(tid * 8 + gg) * 4 + 3];
      if (mm > Mm) { Lm *= __expf(Mm - mm); Mm = mm; }
      Lm += lm * __expf(mm - Mm);
    }
    sLogZ[tid] = Mu + __logf(Lu);
    sMm[tid]   = Mm;
    sLinv[tid] = 1.0f / Lm;
  }
  __syncthreads();

  // ---- Phase 3: emit log_attn & attn; overwrite sS with probabilities -----
  {
    const float logZ = sLogZ[rrow];
    const float mMr  = sMm[rrow];
    const float linv = sLinv[rrow];
    float* oA = outA  + ((size_t)b * LQ_ + q0 + rrow) * LK_ + g * 128;
    float* oL = outLA + ((size_t)b * LQ_ + q0 + rrow) * LK_ + g * 128;
    for (int i = 0; i < 32; ++i) {
      const unsigned int m4 = mrow[i];
#pragma unroll
      for (int j = 0; j < 4; ++j) {
        const float s = srow[i * 4 + j];
        oL[i * 4 + j] = s - logZ;
        const float sm = ((m4 >> (8 * j)) & 0xFFu) ? -1e9f : s;
        const float a  = __expf(sm - mMr) * linv;
        oA[i * 4 + j] = a;
        srow[i * 4 + j] = a;
      }
    }
  }
  __syncthreads();

  // ---- Phase 4: O = P V via WMMA f32 16x16x4 ------------------------------
  // Wave w owns the 16 output dims n0 = w*16; K-loop over 1024 keys.
  {
    const float* vb = vg + (size_t)b * LK_ * D_;
    const int n0 = wave * 16;
    v8f acc0 = {};
    v8f acc1 = {};
    const float* vcol  = vb + (size_t)(half * 2) * D_ + n0 + l15;
    const float* arow0 = sS + (size_t)l15 * LK_ + half * 2;
    const float* arow1 = sS + (size_t)(16 + l15) * LK_ + half * 2;
#pragma unroll 4
    for (int kk = 0; kk < LK_; kk += 4) {
      v2f bv;
      bv.x = vcol[(size_t)kk * D_];          // V[kk + 2*half    ][n]
      bv.y = vcol[(size_t)(kk + 1) * D_];    // V[kk + 2*half + 1][n]
      v2f a0 = *(const v2f*)(arow0 + kk);
      v2f a1 = *(const v2f*)(arow1 + kk);
      acc0 = wmma_f32_4(a0, bv, acc0);
      acc1 = wmma_f32_4(a1, bv, acc1);
    }
    float* ob = outO + ((size_t)b * LQ_ + q0) * D_;
#pragma unroll
    for (int r = 0; r < 8; ++r) {
      const int row = r + half * 8;
      ob[(size_t)row * D_ + n0 + l15]        = acc0[r];
      ob[(size_t)(row + 16) * D_ + n0 + l15] = acc1[r];
    }
  }
}

extern "C" void kernel_launch(void* const* d_in, const int* in_sizes, int n_in,
                              void* d_out, int out_size, void* d_ws, size_t ws_size,
                              hipStream_t stream) {
  (void)in_sizes; (void)n_in; (void)out_size; (void)d_ws; (void)ws_size;
  const float* q = (const float*)d_in[0];
  const float* k = (const float*)d_in[1];
  const float* v = (const float*)d_in[2];
  const unsigned char* mask = (const unsigned char*)d_in[3];  // jnp.bool_ = 1 byte

  float* outO  = (float*)d_out;                               // [B,LQ,D]
  float* outA  = outO + (size_t)B_ * LQ_ * D_;                // [B,LQ,LK]
  float* outLA = outA + (size_t)B_ * LQ_ * LK_;               // [B,LQ,LK]

  const size_t shmem =
      (size_t)(QT * LK_ + QT * D_ + QT * 8 * 4 + 3 * QT) * sizeof(float); // ~148.5KB < 320KB/WGP

  dim3 grid(B_ * (LQ_ / QT));   // 2048 workgroups
  dim3 block(TPB);              // 8 wave32s
  attn_fused_kernel<<<grid, block, shmem, stream>>>(q, k, v, mask, outO, outA, outLA);
}